// _TopKRouter_50646254355258
// MI455X (gfx1250) — compile-verified
//
#include <hip/hip_runtime.h>
#include <hip/hip_bf16.h>

typedef float v2f __attribute__((ext_vector_type(2)));
typedef float v8f __attribute__((ext_vector_type(8)));
typedef int   v4i __attribute__((vector_size(4 * sizeof(int))));
typedef __attribute__((address_space(1))) v4i* gv4i_p;   // global int4*
typedef __attribute__((address_space(3))) v4i* lv4i_p;   // LDS int4*

#define T_TOKENS 16384
#define H_DIM    4096
#define N_EXP    64
#define TOPK     8
#define M_TILE   32
#define BK       64
#define NBLK     (H_DIM / BK)
#define XPITCH   68   // BK + 4-dword pad -> conflict-free 16x4 fragment reads
#define WPITCH   68
#define LPITCH   65

#if __has_builtin(__builtin_amdgcn_global_load_async_to_lds_b128)
#define HAS_ASYNC_LDS 1
#else
#define HAS_ASYNC_LDS 0
#endif

__device__ __forceinline__ void async_cp_b128(const void* g, void* l) {
#if HAS_ASYNC_LDS
    __builtin_amdgcn_global_load_async_to_lds_b128(
        (gv4i_p)(v4i*)(void*)g, (lv4i_p)(v4i*)l, 0, 0);
#endif
}

__device__ __forceinline__ void async_wait_all() {
#if __has_builtin(__builtin_amdgcn_s_wait_asynccnt)
    __builtin_amdgcn_s_wait_asynccnt(0);
#else
    asm volatile("s_wait_asynccnt 0x0" ::: "memory");
#endif
}

__global__ __launch_bounds__(256)
void _TopKRouter_50646254355258_kernel(const float* __restrict__ x,
                                       const float* __restrict__ weight,
                                       const float* __restrict__ bias,
                                       float* __restrict__ out_w,
                                       int* __restrict__ out_i) {
    __shared__ float lds_x[2][M_TILE * XPITCH];
    __shared__ float lds_w[2][N_EXP * WPITCH];
    __shared__ float lds_logits[M_TILE * LPITCH];

    const int tid    = threadIdx.x;
    const int wave   = tid >> 5;
    const int lane   = tid & 31;
    const int lane16 = lane & 15;
    const int hi     = lane >> 4;          // 0 for lanes 0-15, 1 for 16-31
    const int khalf  = hi << 1;            // K offset 0 or 2 within a 16x4 fragment
    const int m_blk  = blockIdx.x * M_TILE;

    const int n_sub = wave & 3;            // expert tile (4 x 16 experts)
    const int m_sub = wave >> 2;           // token tile  (2 x 16 tokens)
    const int a_row = m_sub * 16 + lane16; // token row inside LDS x tile
    const int b_row = n_sub * 16 + lane16; // expert row inside LDS w tile

    // per-thread staging coordinates (16 float4 per 64-col row)
    const int xr0 = tid >> 4,           xc0 = (tid & 15) * 4;          // x: 2 xfers
    const int xr1 = (tid + 256) >> 4,   xc1 = xc0;
    // w: 4 xfers, rows tid>>4 + 16*i
    v8f acc = {};

#if HAS_ASYNC_LDS
    auto stage_async = [&](int blk, int buf) {
        const int kb = blk * BK;
#pragma unroll
        for (int i = 0; i < 2; ++i) {
            int r = (i == 0) ? xr0 : xr1;
            async_cp_b128(x + (size_t)(m_blk + r) * H_DIM + kb + xc0,
                          &lds_x[buf][r * XPITCH + xc0]);
        }
#pragma unroll
        for (int i = 0; i < 4; ++i) {
            int r = (tid >> 4) + 16 * i;
            async_cp_b128(weight + (size_t)r * H_DIM + kb + xc0,
                          &lds_w[buf][r * WPITCH + xc0]);
        }
    };
#else
    float4 xr[2], wr[4];
    auto load_regs = [&](int blk) {
        const int kb = blk * BK;
        xr[0] = *(const float4*)(x + (size_t)(m_blk + xr0) * H_DIM + kb + xc0);
        xr[1] = *(const float4*)(x + (size_t)(m_blk + xr1) * H_DIM + kb + xc1);
#pragma unroll
        for (int i = 0; i < 4; ++i) {
            int r = (tid >> 4) + 16 * i;
            wr[i] = *(const float4*)(weight + (size_t)r * H_DIM + kb + xc0);
        }
    };
    auto store_lds = [&](int buf) {
        {
            float* d0 = &lds_x[buf][xr0 * XPITCH + xc0];
            d0[0] = xr[0].x; d0[1] = xr[0].y; d0[2] = xr[0].z; d0[3] = xr[0].w;
            float* d1 = &lds_x[buf][xr1 * XPITCH + xc1];
            d1[0] = xr[1].x; d1[1] = xr[1].y; d1[2] = xr[1].z; d1[3] = xr[1].w;
        }
#pragma unroll
        for (int i = 0; i < 4; ++i) {
            int r = (tid >> 4) + 16 * i;
            float* d = &lds_w[buf][r * WPITCH + xc0];
            d[0] = wr[i].x; d[1] = wr[i].y; d[2] = wr[i].z; d[3] = wr[i].w;
        }
    };
#endif

    auto compute = [&](int buf) {
#pragma unroll
        for (int kk = 0; kk < BK; kk += 4) {
            v2f a = *(const v2f*)&lds_x[buf][a_row * XPITCH + kk + khalf];
            v2f b = *(const v2f*)&lds_w[buf][b_row * WPITCH + kk + khalf];
            acc = __builtin_amdgcn_wmma_f32_16x16x4_f32(
                false, a, false, b, (short)0, acc, false, false);
        }
    };

    // ---- double-buffered pipeline over K ----
#if HAS_ASYNC_LDS
    stage_async(0, 0);
    async_wait_all();
    __syncthreads();
    for (int ib = 0; ib < NBLK; ++ib) {
        const int cur = ib & 1;
        if (ib + 1 < NBLK) stage_async(ib + 1, cur ^ 1);  // overlap with compute
        compute(cur);
        async_wait_all();
        __syncthreads();
    }
#else
    load_regs(0);
    store_lds(0);
    __syncthreads();
    for (int ib = 0; ib < NBLK; ++ib) {
        const int cur = ib & 1;
        if (ib + 1 < NBLK) load_regs(ib + 1);             // overlap with compute
        compute(cur);
        if (ib + 1 < NBLK) store_lds(cur ^ 1);
        __syncthreads();
    }
#endif

    // ---- scatter C (+bias) into LDS logits tile per documented C layout ----
    const float bv = bias[n_sub * 16 + lane16];
#pragma unroll
    for (int r = 0; r < 8; ++r) {
        int row = m_sub * 16 + r + (hi << 3);   // M = r (lanes 0-15) or r+8
        int col = n_sub * 16 + lane16;          // N = lane
        lds_logits[row * LPITCH + col] = acc[r] + bv;
    }
    __syncthreads();

    // ---- top-8 (stable, lowest index wins ties) + softmax; 1 thread/token ----
    if (tid < M_TILE) {
        const float* lrow = &lds_logits[tid * LPITCH];
        float bvv[TOPK];
        int   bii[TOPK];
#pragma unroll
        for (int i = 0; i < TOPK; ++i) { bvv[i] = -3.402823466e38f; bii[i] = 0; }
        for (int e = 0; e < N_EXP; ++e) {
            float v = lrow[e];
            if (v > bvv[TOPK - 1]) {
                int p = TOPK - 1;
                while (p > 0 && v > bvv[p - 1]) {
                    bvv[p] = bvv[p - 1]; bii[p] = bii[p - 1]; --p;
                }
                bvv[p] = v; bii[p] = e;
            }
        }
        float mx = bvv[0], s = 0.f, ex[TOPK];
#pragma unroll
        for (int i = 0; i < TOPK; ++i) { ex[i] = expf(bvv[i] - mx); s += ex[i]; }
        float inv = 1.0f / s;
        int token = m_blk + tid;
#pragma unroll
        for (int i = 0; i < TOPK; ++i) {
            out_w[token * TOPK + i] = ex[i] * inv;
            out_i[token * TOPK + i] = bii[i];
        }
    }
}

extern "C" void kernel_launch(void* const* d_in, const int* in_sizes, int n_in,
                              void* d_out, int out_size, void* d_ws, size_t ws_size,
                              hipStream_t stream) {
    const float* x      = (const float*)d_in[0];
    const float* weight = (const float*)d_in[1];
    const float* bias   = (const float*)d_in[2];
    float* out_w = (float*)d_out;
    int*   out_i = (int*)((float*)d_out + (size_t)T_TOKENS * TOPK);

    dim3 grid(T_TOKENS / M_TILE);
    _TopKRouter_50646254355258_kernel<<<grid, 256, 0, stream>>>(
        x, weight, bias, out_w, out_i);
}